// LensAutoEncoder_77309411328140
// MI455X (gfx1250) — compile-verified
//
#include <hip/hip_runtime.h>
#include <hip/hip_bf16.h>

#define TOTAL_ELEMS 16777216LL   // B*N*N = 256*256*256
#define NPIX 65536               // N*N
#define CENTER_Q 32896           // q where px==0 && py==0 (128*256+128)

// ---------------------------------------------------------------------------
// Kernel 1: zero the output image and the max-accumulator in workspace.
// ---------------------------------------------------------------------------
__global__ void lens_zero_kernel(float* __restrict__ out, unsigned* __restrict__ ws) {
    long long i = ((long long)blockIdx.x * blockDim.x + threadIdx.x) * 4;
    if (i < TOTAL_ELEMS) {
        *(float4*)(out + i) = make_float4(0.f, 0.f, 0.f, 0.f);
    }
    if (blockIdx.x == 0 && threadIdx.x == 0) ws[0] = 0u;
}

// ---------------------------------------------------------------------------
// Kernel 2: lens scatter.
// 4096 blocks x 256 threads, 4 pipelined iterations, 1024 elems per block-iter.
// x_true is streamed through LDS via the CDNA5 async global->LDS path
// (double-buffered, ASYNCcnt-tracked), overlapping the DMA with the
// sigmoid/lens arithmetic on the directly-loaded x stream. The next x tile is
// software-prefetched (global_prefetch_b8).
// ---------------------------------------------------------------------------
__global__ void lens_scatter_kernel(const float* __restrict__ x,
                                    const float* __restrict__ xt,
                                    float* __restrict__ out) {
    __shared__ float lds_xt[2][1024];   // 2 x 4KB double buffer

    const int tid = threadIdx.x;
    // LDS byte addresses for this thread's 16B slice in each buffer
    const unsigned ldsOff0 = (unsigned)(size_t)(&lds_xt[0][tid * 4]);
    const unsigned ldsOff1 = (unsigned)(size_t)(&lds_xt[1][tid * 4]);

    const long long tileStride = (long long)gridDim.x * 1024;
    const long long base0 = (long long)blockIdx.x * 1024;

    // Prologue: async-load tile 0 of x_true into buffer 0 (GV addressing).
    {
        const float* g = xt + base0 + (long long)tid * 4;
        asm volatile("global_load_async_to_lds_b128 %0, %1, off"
                     :: "v"(ldsOff0), "v"(g) : "memory");
    }

    for (int it = 0; it < 4; ++it) {
        const long long base = base0 + (long long)it * tileStride;

        // Issue the next tile's async load into the other buffer, and
        // prefetch the next x tile, before touching this tile's data.
        if (it < 3) {
            const long long nbase = base + tileStride;
            const float* g = xt + nbase + (long long)tid * 4;
            asm volatile("global_load_async_to_lds_b128 %0, %1, off"
                         :: "v"((it & 1) ? ldsOff0 : ldsOff1), "v"(g) : "memory");
            __builtin_prefetch(x + nbase + (long long)tid * 4, 0, 3);
        }

        // Direct vector load of x for this tile (overlaps with async DMA).
        const float4 xv4 = *(const float4*)(x + base + (long long)tid * 4);
        float xs[4] = {xv4.x, xv4.y, xv4.z, xv4.w};

        // Wait for this tile's async LDS fill; allow the just-issued next
        // tile to stay outstanding (<=1), drain fully on the last iter.
        if (it < 3) {
            asm volatile("s_wait_asynccnt 0x1" ::: "memory");
        } else {
            asm volatile("s_wait_asynccnt 0x0" ::: "memory");
        }

        const float* lb = &lds_xt[it & 1][tid * 4];
        float vals[4] = {lb[0], lb[1], lb[2], lb[3]};

        const long long e0 = base + (long long)tid * 4;
        const int b  = (int)(e0 >> 16);        // batch index
        const int q0 = (int)(e0 & (NPIX - 1)); // flat pixel index in image
        float* ob = out + ((long long)b << 16);

#pragma unroll
        for (int j = 0; j < 4; ++j) {
            const int q = q0 + j;
            if (q == CENTER_Q) continue;       // masked-out center pixel
            const float px = (float)((q & 255) - 128);  // tile(idx,n): fastest dim
            const float py = (float)((q >> 8) - 128);   // repeat(idx,n)
            // k = 0.8 + 0.4*sigmoid(x), accurate exp to match jax.nn.sigmoid
            const float k = 0.8f + 0.4f * (1.0f / (1.0f + expf(-xs[j])));
            const float r = sqrtf(px * px + py * py);
            // match reference op order: p - (k*p)/r
            const float lx = px - (k * px) / r;
            const float ly = py - (k * py) / r;
            // jnp.round == round-half-even == v_rndne
            int ix = (int)rintf(lx + 128.0f);
            int iy = (int)rintf(ly + 128.0f);
            int idx = (ix << 8) + iy;
            idx = min(max(idx, 0), NPIX - 1);
            ob[idx] = vals[j];                 // scatter, last-write-wins
        }
    }
}

// ---------------------------------------------------------------------------
// Kernel 3: global max of out. wave32 shuffle reduce + cross-wave LDS reduce,
// nonnegative-float-as-uint atomicMax into ws[0].
// ---------------------------------------------------------------------------
__global__ void lens_max_kernel(const float* __restrict__ out,
                                unsigned* __restrict__ ws) {
    __shared__ float smax[8];
    const long long stride = (long long)gridDim.x * blockDim.x * 4;
    long long i = ((long long)blockIdx.x * blockDim.x + threadIdx.x) * 4;
    float m = 0.0f;
    for (; i < TOTAL_ELEMS; i += stride) {
        const float4 v = *(const float4*)(out + i);
        m = fmaxf(m, fmaxf(fmaxf(v.x, v.y), fmaxf(v.z, v.w)));
    }
    // wave32 reduction
    for (int off = 16; off > 0; off >>= 1)
        m = fmaxf(m, __shfl_xor(m, off, 32));
    const int wave = threadIdx.x >> 5;
    if ((threadIdx.x & 31) == 0) smax[wave] = m;
    __syncthreads();
    if (threadIdx.x == 0) {
        float bm = smax[0];
#pragma unroll
        for (int w = 1; w < 8; ++w) bm = fmaxf(bm, smax[w]);
        // all values >= 0: uint ordering == float ordering
        atomicMax(ws, __float_as_uint(bm));
    }
}

// ---------------------------------------------------------------------------
// Kernel 4: out /= (max + eps)
// ---------------------------------------------------------------------------
__global__ void lens_norm_kernel(float* __restrict__ out,
                                 const unsigned* __restrict__ ws) {
    const float mx = __uint_as_float(*ws);
    const float d = mx + 1e-9f;
    const long long i = ((long long)blockIdx.x * blockDim.x + threadIdx.x) * 4;
    if (i < TOTAL_ELEMS) {
        float4 v = *(float4*)(out + i);
        v.x = v.x / d; v.y = v.y / d; v.z = v.z / d; v.w = v.w / d;
        *(float4*)(out + i) = v;
    }
}

extern "C" void kernel_launch(void* const* d_in, const int* in_sizes, int n_in,
                              void* d_out, int out_size, void* d_ws, size_t ws_size,
                              hipStream_t stream) {
    const float* x  = (const float*)d_in[0];   // (256,256,256) f32
    const float* xt = (const float*)d_in[1];   // (256,256,256) f32
    float* out = (float*)d_out;                // (256,1,256,256) f32
    unsigned* ws = (unsigned*)d_ws;            // [0]: max accumulator bits

    // 16.7M elems, 4/thread, 256 threads/block -> 16384 blocks
    lens_zero_kernel<<<16384, 256, 0, stream>>>(out, ws);
    // 4096 blocks * 4 iters * 1024 elems = 16.7M
    lens_scatter_kernel<<<4096, 256, 0, stream>>>(x, xt, out);
    lens_max_kernel<<<4096, 256, 0, stream>>>(out, ws);
    lens_norm_kernel<<<16384, 256, 0, stream>>>(out, ws);
}